// GCNNSingleKernel_43026982371861
// MI455X (gfx1250) — compile-verified
//
#include <hip/hip_runtime.h>
#include <hip/hip_bf16.h>
#include <math.h>

// Problem constants (from the reference)
#define B_  16
#define F0_ 16
#define FM_ 96
#define N_  1024
#define H_  48
#define EPS_ 1e-5f

typedef __attribute__((ext_vector_type(2))) float v2f;
typedef __attribute__((ext_vector_type(8))) float v8f;

// D(16x16,f32) = A(16x4,f32) x B(4x16,f32) + C
// Layouts (ISA 7.12.2):
//  A: lane l<16 holds row M=l  : v0=A[l,0], v1=A[l,1];  lane l>=16: v0=A[l-16,2], v1=A[l-16,3]
//  B: lane j<16 holds col N=j  : v0=B[0,j], v1=B[1,j];  lane j>=16: v0=B[2,j-16], v1=B[3,j-16]
//  C/D: VGPR r, lanes 0-15 -> M=r, N=lane; lanes 16-31 -> M=r+8, N=lane-16
static __device__ __forceinline__ v8f wmma4(v2f a, v2f b, v8f c) {
  return __builtin_amdgcn_wmma_f32_16x16x4_f32(
      /*neg_a=*/false, a, /*neg_b=*/false, b,
      /*c_mod=*/(short)0, c, /*reuse_a=*/false, /*reuse_b=*/false);
}

static __device__ __forceinline__ float wave_sum(float s) {
  #pragma unroll
  for (int o = 16; o > 0; o >>= 1) s += __shfl_xor(s, o, 32);
  return s;
}

// ---------------------------------------------------------------- node (diag)
__global__ void node_kernel(const float* __restrict__ adj_mask,
                            float* __restrict__ node) {
  int tid = blockIdx.x * blockDim.x + threadIdx.x;        // over B*N
  int b = tid / N_, n = tid % N_;
  node[tid] = adj_mask[(size_t)b * N_ * N_ + (size_t)n * N_ + n];
}

// ----------------------------------------------------------------- sq = sum f E^2
__global__ void sq_kernel(const float* __restrict__ E, int F,
                          float* __restrict__ sq) {
  int tid = blockIdx.x * blockDim.x + threadIdx.x;        // over B*N
  int b = tid / N_, n = tid % N_;
  const float* Eb = E + (size_t)b * F * N_;
  float s = 0.f;
  for (int f = 0; f < F; ++f) {
    float e = Eb[(size_t)f * N_ + n];
    s += e * e;
  }
  sq[tid] = s;
}

// Epilogue for one 16x16 gram tile -> masked Gaussian adjacency
static __device__ __forceinline__ void adj_epilogue(
    v8f c, int nbase, int mcol, float sm, float nm,
    const float* __restrict__ sqb, const float* __restrict__ ndb,
    float alpha, float* __restrict__ adjb, int half) {
  #pragma unroll
  for (int r = 0; r < 8; ++r) {
    int n = nbase + r + half * 8;
    float d = fmaxf(sqb[n] + sm - 2.0f * c[r], 0.0f);
    adjb[(size_t)n * N_ + mcol] = __expf(-alpha * d) * (ndb[n] * nm);
  }
}

// --------------- adj = exp(-a*dist)*mask, 2x2 register-blocked WMMA gram
// Each wave produces a 32x32 block: 4 loads feed 4 WMMAs per K-step.
__global__ void adj_kernel(const float* __restrict__ E, int F,
                           const float* __restrict__ alphas, int li,
                           const float* __restrict__ sq,
                           const float* __restrict__ node,
                           float* __restrict__ adj) {
  int wid  = blockIdx.x * (blockDim.x >> 5) + (threadIdx.x >> 5);  // tile id
  int lane = threadIdx.x & 31;
  // tiles: b * 32(n super-tiles) * 32(m super-tiles)
  int tj = wid & 31;
  int ti = (wid >> 5) & 31;
  int b  = wid >> 10;
  int n0 = ti * 32, m0 = tj * 32;
  int half = lane >> 4, lid = lane & 15;
  const float* Eb = E + (size_t)b * F * N_;

  v8f c00 = {}, c01 = {}, c10 = {}, c11 = {};
  for (int f0 = 0; f0 < F; f0 += 4) {
    int fk = f0 + 2 * half;
    const float* r0 = Eb + (size_t)(fk + 0) * N_;
    const float* r1 = Eb + (size_t)(fk + 1) * N_;
    v2f a0, a1, b0, b1;
    a0.x = r0[n0 + lid];       a0.y = r1[n0 + lid];
    a1.x = r0[n0 + 16 + lid];  a1.y = r1[n0 + 16 + lid];
    b0.x = r0[m0 + lid];       b0.y = r1[m0 + lid];
    b1.x = r0[m0 + 16 + lid];  b1.y = r1[m0 + 16 + lid];
    c00 = wmma4(a0, b0, c00);
    c01 = wmma4(a0, b1, c01);
    c10 = wmma4(a1, b0, c10);
    c11 = wmma4(a1, b1, c11);
  }

  float alpha = alphas[li];
  const float* sqb = sq + (size_t)b * N_;
  const float* ndb = node + (size_t)b * N_;
  float sm0 = sqb[m0 + lid],      nm0 = ndb[m0 + lid];
  float sm1 = sqb[m0 + 16 + lid], nm1 = ndb[m0 + 16 + lid];
  float* adjb = adj + (size_t)b * N_ * N_;

  adj_epilogue(c00, n0,      m0 + lid,      sm0, nm0, sqb, ndb, alpha, adjb, half);
  adj_epilogue(c01, n0,      m0 + 16 + lid, sm1, nm1, sqb, ndb, alpha, adjb, half);
  adj_epilogue(c10, n0 + 16, m0 + lid,      sm0, nm0, sqb, ndb, alpha, adjb, half);
  adj_epilogue(c11, n0 + 16, m0 + 16 + lid, sm1, nm1, sqb, ndb, alpha, adjb, half);
}

// ----------------------------------------------------------------- deg = row sums
__global__ void deg_kernel(const float* __restrict__ adj,
                           float* __restrict__ deg) {
  int row  = blockIdx.x * (blockDim.x >> 5) + (threadIdx.x >> 5);  // over B*N
  int lane = threadIdx.x & 31;
  const float* p = adj + (size_t)row * N_;
  float s = 0.f;
  for (int m = lane; m < N_; m += 32) s += p[m];
  s = wave_sum(s);
  if (lane == 0) deg[row] = s;
}

// ------------------------------------------- spatial norm (in place) * node
__global__ void snorm_kernel(float* __restrict__ E,
                             const float* __restrict__ node,
                             const int* __restrict__ nb) {
  int wid  = blockIdx.x * (blockDim.x >> 5) + (threadIdx.x >> 5);  // over B*FM
  int lane = threadIdx.x & 31;
  int b = wid / FM_;
  float* row = E + (size_t)wid * N_;
  const float* nd = node + (size_t)b * N_;
  float nbf = (float)nb[b];

  float s = 0.f;
  for (int n = lane; n < N_; n += 32) s += row[n] * nd[n];
  float m = wave_sum(s) / nbf;

  float vs = 0.f;
  for (int n = lane; n < N_; n += 32) {
    float cc = (row[n] - m) * nd[n];
    vs += cc * cc;
  }
  float v = wave_sum(vs) / nbf;
  float inv = 1.0f / sqrtf(v + EPS_);

  for (int n = lane; n < N_; n += 32)
    row[n] = (row[n] - m) * inv * nd[n];
}

// --------- prop[f,n] = sum_m adj[n,m]*E[f,m], FBx2 register-blocked WMMA
// FB = f-tile blocking (2 for F=96, 1 for F=16); always 2 n-tiles per wave.
template <int FB>
__global__ void prop_kernel(const float* __restrict__ adj,
                            const float* __restrict__ E, int F,
                            float* __restrict__ prop) {
  int nfg  = (F >> 4) / FB;                 // f-tile groups
  int wid  = blockIdx.x * (blockDim.x >> 5) + (threadIdx.x >> 5);
  int lane = threadIdx.x & 31;
  int tj = wid % 32;                        // n super-tile (32 cols)
  int fi = (wid / 32) % nfg;
  int b  = wid / (32 * nfg);
  int f0 = fi * 16 * FB, n0 = tj * 32;
  int half = lane >> 4, lid = lane & 15;
  const float* Eb   = E   + (size_t)b * F * N_;
  const float* adjb = adj + (size_t)b * N_ * N_;

  v8f acc[FB][2] = {};
  for (int m0 = 0; m0 < N_; m0 += 4) {
    v2f bt0 = *(const v2f*)&adjb[(size_t)(n0 + lid) * N_ + m0 + 2 * half];
    v2f bt1 = *(const v2f*)&adjb[(size_t)(n0 + 16 + lid) * N_ + m0 + 2 * half];
    #pragma unroll
    for (int k = 0; k < FB; ++k) {
      v2f a = *(const v2f*)&Eb[(size_t)(f0 + 16 * k + lid) * N_ + m0 + 2 * half];
      acc[k][0] = wmma4(a, bt0, acc[k][0]);
      acc[k][1] = wmma4(a, bt1, acc[k][1]);
    }
  }
  float* pb = prop + (size_t)b * F * N_;
  #pragma unroll
  for (int k = 0; k < FB; ++k) {
    #pragma unroll
    for (int r = 0; r < 8; ++r) {
      int f = f0 + 16 * k + r + half * 8;
      pb[(size_t)f * N_ + n0 + lid]      = acc[k][0][r];
      pb[(size_t)f * N_ + n0 + 16 + lid] = acc[k][1][r];
    }
  }
}

// cat row r of [E; E*deg; prop]
static __device__ __forceinline__ float cat_val(const float* __restrict__ E,
                                                const float* __restrict__ P,
                                                float dn, int r, int n, int F) {
  if (r < F)      return E[(size_t)r * N_ + n];
  if (r < 2 * F)  return E[(size_t)(r - F) * N_ + n] * dn;
  return P[(size_t)(r - 2 * F) * N_ + n];
}

// conv = relu(Wc@cat + bc) -> out[0:48); res = Wr@E + br -> out[48:96)
// One wave handles all 3 h-tiles (H=48): the cat/E operand is built once
// per K-step and reused by 3 WMMAs.
__global__ void convres_kernel(const float* __restrict__ Esn,
                               const float* __restrict__ prop,
                               const float* __restrict__ deg,
                               const float* __restrict__ Wc,
                               const float* __restrict__ bc,
                               const float* __restrict__ Wr,
                               const float* __restrict__ br,
                               int F, float* __restrict__ out) {
  int wid  = blockIdx.x * (blockDim.x >> 5) + (threadIdx.x >> 5);
  int lane = threadIdx.x & 31;
  int tj = wid % 64;            // n tile
  int b  = wid / 64;
  int n0 = tj * 16;
  int half = lane >> 4, lid = lane & 15;
  int K3 = 3 * F;
  const float* Eb = Esn  + (size_t)b * F * N_;
  const float* Pb = prop + (size_t)b * F * N_;
  int nn = n0 + lid;
  float dn = deg[(size_t)b * N_ + nn];

  // conv accumulation over cat (3F) for all 3 h-tiles
  v8f cc[3] = {};
  for (int k0 = 0; k0 < K3; k0 += 4) {
    int r0 = k0 + 2 * half;
    v2f bb;
    bb.x = cat_val(Eb, Pb, dn, r0,     nn, F);
    bb.y = cat_val(Eb, Pb, dn, r0 + 1, nn, F);
    #pragma unroll
    for (int hi = 0; hi < 3; ++hi) {
      v2f a = *(const v2f*)&Wc[(size_t)(hi * 16 + lid) * K3 + k0 + 2 * half];
      cc[hi] = wmma4(a, bb, cc[hi]);
    }
  }
  // residual accumulation over F for all 3 h-tiles
  v8f rr[3] = {};
  for (int k0 = 0; k0 < F; k0 += 4) {
    int r0 = k0 + 2 * half;
    v2f bb;
    bb.x = Eb[(size_t)(r0)     * N_ + nn];
    bb.y = Eb[(size_t)(r0 + 1) * N_ + nn];
    #pragma unroll
    for (int hi = 0; hi < 3; ++hi) {
      v2f a = *(const v2f*)&Wr[(size_t)(hi * 16 + lid) * F + k0 + 2 * half];
      rr[hi] = wmma4(a, bb, rr[hi]);
    }
  }

  float* ob = out + (size_t)b * FM_ * N_;
  #pragma unroll
  for (int hi = 0; hi < 3; ++hi) {
    #pragma unroll
    for (int q = 0; q < 8; ++q) {
      int h = hi * 16 + q + half * 8;
      float cv = fmaxf(cc[hi][q] + bc[h], 0.0f);
      float rv = rr[hi][q] + br[h];
      ob[(size_t)h * N_ + nn]        = cv;   // conv half
      ob[(size_t)(H_ + h) * N_ + nn] = rv;   // residual half
    }
  }
}

// ----------------------- masked pool, feature-normalize, fc, sigmoid -> out[b]
__global__ void final_kernel(const float* __restrict__ E,
                             const float* __restrict__ node,
                             const float* __restrict__ fclw,
                             const float* __restrict__ fclb,
                             float* __restrict__ out) {
  int b = blockIdx.x;
  __shared__ float pooled[FM_];
  const float* Eb = E + (size_t)b * FM_ * N_;
  const float* nd = node + (size_t)b * N_;
  int wv = threadIdx.x >> 5, lane = threadIdx.x & 31;

  for (int f = wv; f < FM_; f += 4) {
    float s = 0.f;
    for (int n = lane; n < N_; n += 32) s += Eb[(size_t)f * N_ + n] * nd[n];
    s = wave_sum(s);
    if (lane == 0) pooled[f] = s;
  }
  __syncthreads();

  if (wv == 0) {
    float s = 0.f;
    for (int f = lane; f < FM_; f += 32) s += pooled[f];
    float m = wave_sum(s) / (float)FM_;
    float s2 = 0.f;
    for (int f = lane; f < FM_; f += 32) {
      float d = pooled[f] - m;
      s2 += d * d;
    }
    float v = wave_sum(s2) / (float)FM_;
    float inv = 1.0f / sqrtf(v + EPS_);
    float dot = 0.f;
    for (int f = lane; f < FM_; f += 32)
      dot += (pooled[f] - m) * inv * fclw[f];
    dot = wave_sum(dot);
    if (lane == 0) out[b] = 1.0f / (1.0f + __expf(-(dot + fclb[0])));
  }
}

extern "C" void kernel_launch(void* const* d_in, const int* in_sizes, int n_in,
                              void* d_out, int out_size, void* d_ws, size_t ws_size,
                              hipStream_t stream) {
  const float* emb_in   = (const float*)d_in[0];
  const float* adj_mask = (const float*)d_in[1];
  const int*   nb       = (const int*)  d_in[2];
  const float* alphas   = (const float*)d_in[3];
  const float* Wc0      = (const float*)d_in[4];
  const float* bc0      = (const float*)d_in[5];
  const float* Wr0      = (const float*)d_in[6];
  const float* br0      = (const float*)d_in[7];
  const float* WcL      = (const float*)d_in[8];
  const float* bcL      = (const float*)d_in[9];
  const float* WrL      = (const float*)d_in[10];
  const float* brL      = (const float*)d_in[11];
  const float* fclw     = (const float*)d_in[12];
  const float* fclb     = (const float*)d_in[13];
  float* out = (float*)d_out;

  float* ws = (float*)d_ws;
  size_t o = 0;
  float* adj  = ws + o; o += (size_t)B_ * N_ * N_;   // 64 MB
  float* embA = ws + o; o += (size_t)B_ * FM_ * N_;  // 6 MB
  float* embB = ws + o; o += (size_t)B_ * FM_ * N_;  // 6 MB
  float* prop = ws + o; o += (size_t)B_ * FM_ * N_;  // 6 MB
  float* sq   = ws + o; o += (size_t)B_ * N_;
  float* node = ws + o; o += (size_t)B_ * N_;
  float* deg  = ws + o; o += (size_t)B_ * N_;

  node_kernel<<<(B_ * N_) / 256, 256, 0, stream>>>(adj_mask, node);

  // ---- layer 0 (F = 16, no spatial norm)
  sq_kernel<<<(B_ * N_) / 256, 256, 0, stream>>>(emb_in, F0_, sq);
  adj_kernel<<<(B_ * 32 * 32) / 4, 128, 0, stream>>>(emb_in, F0_, alphas, 0, sq, node, adj);
  deg_kernel<<<(B_ * N_) / 8, 256, 0, stream>>>(adj, deg);
  prop_kernel<1><<<(B_ * 1 * 32) / 4, 128, 0, stream>>>(adj, emb_in, F0_, prop);
  convres_kernel<<<(B_ * 64) / 4, 128, 0, stream>>>(emb_in, prop, deg,
                                                    Wc0, bc0, Wr0, br0, F0_, embA);

  // ---- layers 1..3 (F = 96)
  float* cur = embA;
  float* nxt = embB;
  for (int i = 0; i < 3; ++i) {
    sq_kernel<<<(B_ * N_) / 256, 256, 0, stream>>>(cur, FM_, sq);
    adj_kernel<<<(B_ * 32 * 32) / 4, 128, 0, stream>>>(cur, FM_, alphas, i + 1, sq, node, adj);
    deg_kernel<<<(B_ * N_) / 8, 256, 0, stream>>>(adj, deg);
    snorm_kernel<<<(B_ * FM_) / 8, 256, 0, stream>>>(cur, node, nb);
    prop_kernel<2><<<(B_ * 3 * 32) / 4, 128, 0, stream>>>(adj, cur, FM_, prop);
    convres_kernel<<<(B_ * 64) / 4, 128, 0, stream>>>(cur, prop, deg,
                                                      WcL + (size_t)i * H_ * 3 * FM_,
                                                      bcL + (size_t)i * H_,
                                                      WrL + (size_t)i * H_ * FM_,
                                                      brL + (size_t)i * H_,
                                                      FM_, nxt);
    float* t = cur; cur = nxt; nxt = t;
  }

  final_kernel<<<B_, 128, 0, stream>>>(cur, node, fclw, fclb, out);
}